// FlowSmoothLoss_72361609003693
// MI455X (gfx1250) — compile-verified
//
#include <hip/hip_runtime.h>
#include <math.h>

typedef __attribute__((ext_vector_type(2))) float v2f;
typedef __attribute__((ext_vector_type(8))) float v8f;

#define BSZ 4
#define NPT 100000
#define KMSK 16

// workspace layout in 32-bit words
#define OFF_HIST_HI 0u
#define OFF_HIST_L0 (OFF_HIST_HI + BSZ*65536u)
#define OFF_HIST_L1 (OFF_HIST_L0 + BSZ*65536u)
#define OFF_SELINFO (OFF_HIST_L1 + BSZ*65536u)   // BSZ*4 u32: hi0,pre0,hi1,pre1
#define OFF_THR     (OFF_SELINFO + BSZ*4u)       // BSZ f32
#define OFF_STATS   (OFF_THR + BSZ)              // BSZ*7 f32: cnt,sx,sy,sz,sxx,syy,szz
#define OFF_PARAMS  (OFF_STATS + BSZ*7u)         // BSZ*2 f32: s, 1/n
#define OFF_GRAM    (OFF_PARAMS + BSZ*2u)        // BSZ*KMSK*28 f32 (4x7 blocks)
#define OFF_THETA   (OFF_GRAM + BSZ*KMSK*28u)    // BSZ*KMSK*12 f32
#define OFF_LACC    (OFF_THETA + BSZ*KMSK*12u)   // BSZ*2 f32: each, sum
#define WS_WORDS    (OFF_LACC + BSZ*2u)

#define GB 128
#define NSPLIT 64
#define LB 128

#if __has_builtin(__builtin_amdgcn_global_load_async_to_lds_b32)
#define HAVE_ASYNC_LDS 1
#else
#define HAVE_ASYNC_LDS 0
#endif

typedef __attribute__((address_space(1))) int* gas_i32p;
typedef __attribute__((address_space(3))) int* las_i32p;

__device__ __forceinline__ void wait_asynccnt0() {
#if __has_builtin(__builtin_amdgcn_s_wait_asynccnt)
  __builtin_amdgcn_s_wait_asynccnt(0);
#else
  asm volatile("s_wait_asynccnt 0" ::: "memory");
#endif
}

// ---------- pass 1: histogram of high 16 bits of float(|flow|) ----------
__global__ __launch_bounds__(256) void k_hist_hi(const float* __restrict__ flow,
                                                 unsigned* __restrict__ wsU) {
  int b = blockIdx.x / GB, blk = blockIdx.x % GB;
  unsigned* hist = wsU + OFF_HIST_HI + (unsigned)b * 65536u;
  for (int n = blk * 256 + threadIdx.x; n < NPT; n += GB * 256) {
    const float* f = flow + ((size_t)b * NPT + n) * 3;
    float mag = sqrtf(f[0]*f[0] + f[1]*f[1] + f[2]*f[2]);
    unsigned bits = __float_as_uint(mag);
    atomicAdd(&hist[bits >> 16], 1u);
  }
}

// cooperative "find bin holding rank-th element" over a 65536-bin histogram
__device__ __forceinline__ void find_kth_bin(const unsigned* __restrict__ hist,
                                             unsigned rank, unsigned* sm,
                                             unsigned& bin, unsigned& pre) {
  int t = threadIdx.x;
  unsigned ssum = 0;
  for (int i = 0; i < 256; ++i) ssum += hist[t * 256 + i];
  sm[t] = ssum;
  __syncthreads();
  if (t == 0) {
    unsigned cum = 0; int seg = 255;
    for (int i = 0; i < 256; ++i) {
      unsigned c = sm[i];
      if (cum + c > rank) { seg = i; break; }
      cum += c;
    }
    unsigned cum2 = cum; int bb = seg * 256 + 255;
    for (int i = seg * 256; i < seg * 256 + 256; ++i) {
      unsigned c = hist[i];
      if (cum2 + c > rank) { bb = i; break; }
      cum2 += c;
    }
    sm[256] = (unsigned)bb; sm[257] = cum2;
  }
  __syncthreads();
  bin = sm[256]; pre = sm[257];
  __syncthreads();
}

__global__ __launch_bounds__(256) void k_find_hi(unsigned* __restrict__ wsU,
                                                 unsigned r0, unsigned r1) {
  __shared__ unsigned sm[258];
  int b = blockIdx.x;
  const unsigned* hist = wsU + OFF_HIST_HI + (unsigned)b * 65536u;
  unsigned b0, p0, b1, p1;
  find_kth_bin(hist, r0, sm, b0, p0);
  find_kth_bin(hist, r1, sm, b1, p1);
  if (threadIdx.x == 0) {
    unsigned* si = wsU + OFF_SELINFO + (unsigned)b * 4u;
    si[0] = b0; si[1] = p0; si[2] = b1; si[3] = p1;
  }
}

__global__ __launch_bounds__(256) void k_hist_lo(const float* __restrict__ flow,
                                                 unsigned* __restrict__ wsU) {
  int b = blockIdx.x / GB, blk = blockIdx.x % GB;
  const unsigned* si = wsU + OFF_SELINFO + (unsigned)b * 4u;
  unsigned hi0 = si[0], hi1 = si[2];
  unsigned* h0 = wsU + OFF_HIST_L0 + (unsigned)b * 65536u;
  unsigned* h1 = wsU + OFF_HIST_L1 + (unsigned)b * 65536u;
  for (int n = blk * 256 + threadIdx.x; n < NPT; n += GB * 256) {
    const float* f = flow + ((size_t)b * NPT + n) * 3;
    float mag = sqrtf(f[0]*f[0] + f[1]*f[1] + f[2]*f[2]);
    unsigned bits = __float_as_uint(mag);
    unsigned hi = bits >> 16, lo = bits & 0xffffu;
    if (hi == hi0) atomicAdd(&h0[lo], 1u);
    if (hi == hi1) atomicAdd(&h1[lo], 1u);
  }
}

__global__ __launch_bounds__(256) void k_find_lo(unsigned* __restrict__ wsU,
                                                 unsigned r0, unsigned r1, float frac) {
  __shared__ unsigned sm[258];
  int b = blockIdx.x;
  const unsigned* si = wsU + OFF_SELINFO + (unsigned)b * 4u;
  unsigned hi0 = si[0], pre0 = si[1], hi1 = si[2], pre1 = si[3];
  unsigned lo0, q0, lo1, q1;
  find_kth_bin(wsU + OFF_HIST_L0 + (unsigned)b * 65536u, r0 - pre0, sm, lo0, q0);
  find_kth_bin(wsU + OFF_HIST_L1 + (unsigned)b * 65536u, r1 - pre1, sm, lo1, q1);
  if (threadIdx.x == 0) {
    float v0 = __uint_as_float((hi0 << 16) | lo0);
    float v1 = __uint_as_float((hi1 << 16) | lo1);
    ((float*)wsU)[OFF_THR + b] = v0 + frac * (v1 - v0);
  }
}

// ---------- stats: n_sel, sum f, sum f^2 over selected points ----------
__global__ __launch_bounds__(256) void k_stats(const float* __restrict__ flow,
                                               float* __restrict__ wsF) {
  __shared__ float red[256];
  int b = blockIdx.x / GB, blk = blockIdx.x % GB;
  float thr = wsF[OFF_THR + b];
  float acc[7] = {0.f, 0.f, 0.f, 0.f, 0.f, 0.f, 0.f};
  for (int n = blk * 256 + threadIdx.x; n < NPT; n += GB * 256) {
    const float* f = flow + ((size_t)b * NPT + n) * 3;
    float fx = f[0], fy = f[1], fz = f[2];
    float mag = sqrtf(fx*fx + fy*fy + fz*fz);
    if (mag >= thr) {
      acc[0] += 1.f;
      acc[1] += fx; acc[2] += fy; acc[3] += fz;
      acc[4] += fx*fx; acc[5] += fy*fy; acc[6] += fz*fz;
    }
  }
  for (int v = 0; v < 7; ++v) {
    red[threadIdx.x] = acc[v];
    __syncthreads();
    for (int s = 128; s > 0; s >>= 1) {
      if (threadIdx.x < (unsigned)s) red[threadIdx.x] += red[threadIdx.x + s];
      __syncthreads();
    }
    if (threadIdx.x == 0) atomicAdd(&wsF[OFF_STATS + b * 7 + v], red[0]);
    __syncthreads();
  }
}

__global__ void k_params(float* __restrict__ wsF) {
  int b = threadIdx.x;
  if (b < BSZ) {
    const float* st = wsF + OFF_STATS + b * 7;
    float n = st[0];
    float vmax = 0.f;
    for (int d = 0; d < 3; ++d) {
      float mean = st[1 + d] / n;
      float var = (st[4 + d] - n * mean * mean) / fmaxf(n - 1.f, 1.f);
      vmax = fmaxf(vmax, var);
    }
    float s = 10.f / (sqrtf(vmax) + 0.1f);
    wsF[OFF_PARAMS + b * 2]     = s;
    wsF[OFF_PARAMS + b * 2 + 1] = 1.f / n;
  }
}

// ---------- Gram accumulation via V_WMMA_F32_16X16X4_F32 ----------
// Per point: V = (mask*sel) * [x,y,z,1, s*fx, s*fy, s*fz]  (7-vector)
// G_k = sum_n V V^T : rows 0..3 x cols 0..6 hold EtE (4x4) | EtF (4x3).
// Two masks per WMMA: A rows 0..6 = mask k0 comps, rows 8..14 = mask k1 comps.
// B = A^T; with the documented f32 layouts the per-lane registers coincide.
// Mask tile staged through GLOBAL_LOAD_ASYNC_TO_LDS (ASYNCcnt) when available.
__global__ __launch_bounds__(256) void k_gram(const float* __restrict__ pos,
                                              const float* __restrict__ mask,
                                              const float* __restrict__ flow,
                                              float* __restrict__ wsF) {
  __shared__ float wld[256][8];        // V components (sans mask), col 7 = 0 pad
  __shared__ float mld[KMSK][256];     // per-mask tile
  int b = blockIdx.x / NSPLIT, split = blockIdx.x % NSPLIT;
  const int chunk = (NPT + NSPLIT - 1) / NSPLIT;
  int begin = split * chunk;
  int end = begin + chunk; if (end > NPT) end = NPT;

  float thr = wsF[OFF_THR + b];
  float s   = wsF[OFF_PARAMS + b * 2];

  int tid  = threadIdx.x;
  int lane = tid & 31;
  int wave = tid >> 5;                 // 8 waves -> 8 mask pairs
  int idx  = lane & 15;
  int hi   = lane >> 4;                // selects K={0,1} vs {2,3} pair
  int kk   = (idx < 8) ? (2 * wave) : (2 * wave + 1);
  int comp = (idx < 7) ? idx : ((idx >= 8 && idx < 15) ? (idx - 8) : 7);

  v8f acc0 = {};
  v8f acc1 = {};

  for (int p = begin; p < end; p += 256) {
    {
      int i = tid;
      int n = p + i;
      bool valid = (n < end);
      // kick off the mask tile transfer first (dominant stream, 16 rows)
      if (valid) {
#if HAVE_ASYNC_LDS
#pragma unroll
        for (int k = 0; k < KMSK; ++k) {
          const float* gsrc = mask + ((size_t)b * KMSK + k) * NPT + n;
          __builtin_amdgcn_global_load_async_to_lds_b32(
              (gas_i32p)gsrc, (las_i32p)&mld[k][i], 0, 0);
        }
#else
#pragma unroll
        for (int k = 0; k < KMSK; ++k)
          mld[k][i] = mask[((size_t)b * KMSK + k) * NPT + n];
#endif
      } else {
#pragma unroll
        for (int k = 0; k < KMSK; ++k) mld[k][i] = 0.f;
      }
      float x = 0.f, y = 0.f, z = 0.f, fx = 0.f, fy = 0.f, fz = 0.f, sel = 0.f;
      if (valid) {
        const float* pp = pos  + ((size_t)b * NPT + n) * 3;
        const float* ff = flow + ((size_t)b * NPT + n) * 3;
        x = pp[0]; y = pp[1]; z = pp[2];
        fx = ff[0]; fy = ff[1]; fz = ff[2];
        float mag = sqrtf(fx*fx + fy*fy + fz*fz);
        sel = (mag >= thr) ? 1.f : 0.f;
      }
      wld[i][0] = sel * x;  wld[i][1] = sel * y;  wld[i][2] = sel * z;  wld[i][3] = sel;
      wld[i][4] = sel * s * fx; wld[i][5] = sel * s * fy; wld[i][6] = sel * s * fz;
      wld[i][7] = 0.f;
    }
#if HAVE_ASYNC_LDS
    wait_asynccnt0();
#endif
    __syncthreads();
    for (int q = 0; q < 256; q += 8) {
      int p0 = q + 2 * hi;
      v2f av = { mld[kk][p0] * wld[p0][comp], mld[kk][p0 + 1] * wld[p0 + 1][comp] };
      acc0 = __builtin_amdgcn_wmma_f32_16x16x4_f32(false, av, false, av,
                                                   (short)0, acc0, false, false);
      int p1 = q + 4 + 2 * hi;
      v2f bv = { mld[kk][p1] * wld[p1][comp], mld[kk][p1 + 1] * wld[p1 + 1][comp] };
      acc1 = __builtin_amdgcn_wmma_f32_16x16x4_f32(false, bv, false, bv,
                                                   (short)0, acc1, false, false);
    }
    __syncthreads();
  }

  v8f acc = acc0 + acc1;
  float* Gp = wsF + OFF_GRAM;
  if (lane < 7) {                      // mask k0: M=0..3, N=lane
    int base = ((b * KMSK + 2 * wave) * 4) * 7 + lane;
#pragma unroll
    for (int r = 0; r < 4; ++r) atomicAdd(&Gp[base + r * 7], acc[r]);
  } else if (lane >= 24 && lane < 31) { // mask k1: M=8..11 -> acc[0..3], N=8..14
    int col = lane - 24;
    int base = ((b * KMSK + 2 * wave + 1) * 4) * 7 + col;
#pragma unroll
    for (int r = 0; r < 4; ++r) atomicAdd(&Gp[base + r * 7], acc[r]);
  }
}

// ---------- 4x4 pivoted Gauss-Jordan per (b,k) ----------
__global__ void k_solve(float* __restrict__ wsF) {
  int b = blockIdx.x;
  int k = threadIdx.x;
  if (k >= KMSK) return;
  const float* Gp = wsF + OFF_GRAM + (size_t)(b * KMSK + k) * 28;
  float M[4][7];
  for (int r = 0; r < 4; ++r)
    for (int c = 0; c < 7; ++c) M[r][c] = Gp[r * 7 + c];
  for (int c = 0; c < 4; ++c) {
    int piv = c; float best = fabsf(M[c][c]);
    for (int r = c + 1; r < 4; ++r) {
      float a = fabsf(M[r][c]);
      if (a > best) { best = a; piv = r; }
    }
    if (piv != c)
      for (int cc = 0; cc < 7; ++cc) { float t = M[c][cc]; M[c][cc] = M[piv][cc]; M[piv][cc] = t; }
    float inv = 1.f / M[c][c];
    for (int cc = 0; cc < 7; ++cc) M[c][cc] *= inv;
    for (int r = 0; r < 4; ++r)
      if (r != c) {
        float f = M[r][c];
        for (int cc = 0; cc < 7; ++cc) M[r][cc] -= f * M[c][cc];
      }
  }
  float* th = wsF + OFF_THETA + (size_t)(b * KMSK + k) * 12;
  for (int a = 0; a < 4; ++a)
    for (int j = 0; j < 3; ++j) th[a * 3 + j] = M[a][4 + j];
}

// ---------- streaming loss pass ----------
__global__ __launch_bounds__(256) void k_loss(const float* __restrict__ pos,
                                              const float* __restrict__ mask,
                                              const float* __restrict__ flow,
                                              float* __restrict__ wsF) {
  __shared__ float th[KMSK * 12];
  __shared__ float red[256];
  int b = blockIdx.x / LB, blk = blockIdx.x % LB;
  if (threadIdx.x < KMSK * 12)
    th[threadIdx.x] = wsF[OFF_THETA + b * KMSK * 12 + threadIdx.x];
  __syncthreads();
  float thr = wsF[OFF_THR + b];
  float s   = wsF[OFF_PARAMS + b * 2];
  float ea = 0.f, su = 0.f;
  for (int n = blk * 256 + threadIdx.x; n < NPT; n += LB * 256) {
    const float* pp = pos  + ((size_t)b * NPT + n) * 3;
    const float* ff = flow + ((size_t)b * NPT + n) * 3;
    float x = pp[0], y = pp[1], z = pp[2];
    float fx = ff[0], fy = ff[1], fz = ff[2];
    float mag = sqrtf(fx*fx + fy*fy + fz*fz);
    float sel = (mag >= thr) ? 1.f : 0.f;
    float fsx = s * fx, fsy = s * fy, fsz = s * fz;   // fl_n / sel
    float rx = 0.f, ry = 0.f, rz = 0.f;
    for (int k = 0; k < KMSK; ++k) {
      float m = mask[((size_t)b * KMSK + k) * NPT + n] * sel;
      const float* t = &th[k * 12];
      float px = m * (x * t[0] + y * t[3] + z * t[6] + t[9]);
      float py = m * (x * t[1] + y * t[4] + z * t[7] + t[10]);
      float pz = m * (x * t[2] + y * t[5] + z * t[8] + t[11]);
      float dx = px - m * fsx, dy = py - m * fsy, dz = pz - m * fsz;
      ea += dx * dx + dy * dy + dz * dz;
      rx += px; ry += py; rz += pz;
    }
    float ex = sel * fsx - rx, ey = sel * fsy - ry, ez = sel * fsz - rz;
    su += ex * ex + ey * ey + ez * ez;
  }
  ea *= (1.f / 3.f);   // mean over the 3 flow components
  red[threadIdx.x] = ea;
  __syncthreads();
  for (int ss = 128; ss > 0; ss >>= 1) {
    if (threadIdx.x < (unsigned)ss) red[threadIdx.x] += red[threadIdx.x + ss];
    __syncthreads();
  }
  if (threadIdx.x == 0) atomicAdd(&wsF[OFF_LACC + b * 2], red[0]);
  __syncthreads();
  red[threadIdx.x] = su;
  __syncthreads();
  for (int ss = 128; ss > 0; ss >>= 1) {
    if (threadIdx.x < (unsigned)ss) red[threadIdx.x] += red[threadIdx.x + ss];
    __syncthreads();
  }
  if (threadIdx.x == 0) atomicAdd(&wsF[OFF_LACC + b * 2 + 1], red[0]);
}

__global__ void k_final(const float* __restrict__ wsF, float* __restrict__ out) {
  if (threadIdx.x == 0 && blockIdx.x == 0) {
    float tot = 0.f;
    for (int b = 0; b < BSZ; ++b) {
      float n  = wsF[OFF_STATS + b * 7];
      float ea = wsF[OFF_LACC + b * 2];
      float su = wsF[OFF_LACC + b * 2 + 1];
      tot += (0.5f * ea + 0.5f * su) / n;
    }
    out[0] = tot / (float)BSZ;
  }
}

extern "C" void kernel_launch(void* const* d_in, const int* in_sizes, int n_in,
                              void* d_out, int out_size, void* d_ws, size_t ws_size,
                              hipStream_t stream) {
  (void)in_sizes; (void)n_in; (void)out_size; (void)ws_size;
  const float* pos  = (const float*)d_in[0];
  const float* mask = (const float*)d_in[1];
  const float* flow = (const float*)d_in[2];
  unsigned* wsU = (unsigned*)d_ws;
  float*    wsF = (float*)d_ws;

  (void)hipMemsetAsync(d_ws, 0, (size_t)WS_WORDS * 4u, stream);

  double qidx = 0.9 * (double)(NPT - 1);
  unsigned r0 = (unsigned)qidx;
  unsigned r1 = r0 + 1;
  float frac = (float)(qidx - (double)r0);

  k_hist_hi<<<dim3(BSZ * GB), dim3(256), 0, stream>>>(flow, wsU);
  k_find_hi<<<dim3(BSZ), dim3(256), 0, stream>>>(wsU, r0, r1);
  k_hist_lo<<<dim3(BSZ * GB), dim3(256), 0, stream>>>(flow, wsU);
  k_find_lo<<<dim3(BSZ), dim3(256), 0, stream>>>(wsU, r0, r1, frac);
  k_stats<<<dim3(BSZ * GB), dim3(256), 0, stream>>>(flow, wsF);
  k_params<<<dim3(1), dim3(64), 0, stream>>>(wsF);
  k_gram<<<dim3(BSZ * NSPLIT), dim3(256), 0, stream>>>(pos, mask, flow, wsF);
  k_solve<<<dim3(BSZ), dim3(32), 0, stream>>>(wsF);
  k_loss<<<dim3(BSZ * LB), dim3(256), 0, stream>>>(pos, mask, flow, wsF);
  k_final<<<dim3(1), dim3(32), 0, stream>>>(wsF, (float*)d_out);
}